// DetectMultiImage_36687610642990
// MI455X (gfx1250) — compile-verified
//
#include <hip/hip_runtime.h>
#include <cstdint>

// Problem constants (fixed by the reference's setup_inputs()).
#define NN 64
#define CC 18            // NUM_ANCHORS * 6
#define HH 160
#define WW 160
#define AA 3
#define HWPIX (HH * WW)                       // 25600
#define PIX_PER_BLOCK 256
#define ROWS_PER_BLOCK (PIX_PER_BLOCK * AA)   // 768
#define PB_PER_IMG (HWPIX / PIX_PER_BLOCK)    // 100 (exact)
#define NB (NN * PB_PER_IMG)                  // 6400 blocks
#define SCAN_PER_THREAD (NB / 256)            // 25 (exact)
#define CELLF 32.0f
#define ANCHOR_WF 85.72f
#define ANCHOR_HF 19.15f
#define THETA_MARGINF 60.0f                   // 180 / 3

__device__ __forceinline__ float sigmoidf_(float x) {
    return 1.0f / (1.0f + expf(-x));
}

// ---------------------------------------------------------------------------
// Kernel 1: per-block count of valid rows (reads only the 3 confidence planes)
// ---------------------------------------------------------------------------
__global__ void __launch_bounds__(PIX_PER_BLOCK)
count_kernel(const float* __restrict__ in, const float* __restrict__ thr_p,
             int* __restrict__ counts) {
    const int b = blockIdx.x;
    const int t = threadIdx.x;
    const int n = b / PB_PER_IMG;
    const int q = (b % PB_PER_IMG) * PIX_PER_BLOCK + t;   // pixel within image
    const float thr = thr_p[0];

    const float* base = in + (size_t)n * CC * HWPIX + q;
    int cnt = 0;
#pragma unroll
    for (int a = 0; a < AA; ++a) {
        float x0 = base[(size_t)(a * 6) * HWPIX];
        cnt += (sigmoidf_(x0) >= thr) ? 1 : 0;
    }

    __shared__ int red[PIX_PER_BLOCK];
    red[t] = cnt;
    __syncthreads();
    for (int s = PIX_PER_BLOCK / 2; s > 0; s >>= 1) {
        if (t < s) red[t] += red[t + s];
        __syncthreads();
    }
    if (t == 0) counts[b] = red[0];
}

// ---------------------------------------------------------------------------
// Kernel 2: exclusive scan of the 6400 block counts (single workgroup)
// ---------------------------------------------------------------------------
__global__ void __launch_bounds__(256)
scan_kernel(const int* __restrict__ counts, int* __restrict__ offsets,
            int* __restrict__ total) {
    const int t = threadIdx.x;
    int local[SCAN_PER_THREAD];
    int s = 0;
#pragma unroll
    for (int i = 0; i < SCAN_PER_THREAD; ++i) {
        local[i] = counts[t * SCAN_PER_THREAD + i];
        s += local[i];
    }

    __shared__ int sums[256];
    sums[t] = s;
    __syncthreads();
    // Hillis-Steele inclusive scan over 256 partials
    for (int off = 1; off < 256; off <<= 1) {
        int v = sums[t];
        int add = (t >= off) ? sums[t - off] : 0;
        __syncthreads();
        sums[t] = v + add;
        __syncthreads();
    }
    int excl = (t == 0) ? 0 : sums[t - 1];
#pragma unroll
    for (int i = 0; i < SCAN_PER_THREAD; ++i) {
        offsets[t * SCAN_PER_THREAD + i] = excl;
        excl += local[i];
    }
    if (t == 255) total[0] = excl;   // grand total K
}

// ---------------------------------------------------------------------------
// Kernel 3: async global->LDS tile load, decode, compact into LDS,
//           async LDS->global drain of the two contiguous output spans.
// ---------------------------------------------------------------------------
__global__ void __launch_bounds__(PIX_PER_BLOCK)
decode_kernel(const float* __restrict__ in, const float* __restrict__ thr_p,
              const int* __restrict__ offsets, const int* __restrict__ total,
              float* __restrict__ out) {
    const int b = blockIdx.x;
    const int t = threadIdx.x;
    const int n = b / PB_PER_IMG;
    const int q = (b % PB_PER_IMG) * PIX_PER_BLOCK + t;   // pixel within image

    // 18 KB: input tile [c][pixel]; reused afterwards as the compacted
    // output stage: 768 rows x 6 floats == 18*256 floats exactly.
    __shared__ float tile[CC * PIX_PER_BLOCK];
    __shared__ int scan[PIX_PER_BLOCK];

    // --- CDNA5 async global->LDS staging (ASYNCcnt-tracked) -----------------
    {
        const float* imgBase = in + (size_t)n * CC * HWPIX;   // uniform -> SGPR pair
        unsigned ldsAddr = (unsigned)(uintptr_t)(&tile[t]);   // LDS byte addr (c=0 col t)
        unsigned gOff    = (unsigned)q * 4u;                  // byte offset in image
#pragma unroll
        for (int c = 0; c < CC; ++c) {
            unsigned la = ldsAddr + (unsigned)(c * PIX_PER_BLOCK) * 4u;
            unsigned ga = gOff + (unsigned)(c * HWPIX) * 4u;
            asm volatile("global_load_async_to_lds_b32 %0, %1, %2"
                         :
                         : "v"(la), "v"(ga), "s"(imgBase)
                         : "memory");
        }
        asm volatile("s_wait_asynccnt 0x0" ::: "memory");
    }
    __syncthreads();

    const float thr = thr_p[0];
    const int blockOff = offsets[b];   // uniform address -> s_load
    const int K = total[0];            // uniform address -> s_load

    const int iy = q / WW;
    const int ix = q % WW;

    float conf[AA], cx[AA], cy[AA], wv[AA], hv[AA], tv[AA];
    int flag[AA];
    int cnt = 0;
#pragma unroll
    for (int a = 0; a < AA; ++a) {
        float x0 = tile[(a * 6 + 0) * PIX_PER_BLOCK + t];
        float x1 = tile[(a * 6 + 1) * PIX_PER_BLOCK + t];
        float x2 = tile[(a * 6 + 2) * PIX_PER_BLOCK + t];
        float x3 = tile[(a * 6 + 3) * PIX_PER_BLOCK + t];
        float x4 = tile[(a * 6 + 4) * PIX_PER_BLOCK + t];
        float x5 = tile[(a * 6 + 5) * PIX_PER_BLOCK + t];
        conf[a] = sigmoidf_(x0);
        flag[a] = (conf[a] >= thr) ? 1 : 0;
        cx[a] = ((float)ix + sigmoidf_(x1)) * CELLF;
        cy[a] = ((float)iy + sigmoidf_(x2)) * CELLF;
        wv[a] = ANCHOR_WF * expf(x3);
        hv[a] = ANCHOR_HF * expf(x4);
        tv[a] = ((float)a + sigmoidf_(x5)) * THETA_MARGINF;
        cnt += flag[a];
    }

    // Intra-block exclusive scan of per-thread valid counts (rows are ordered
    // thread-major, anchor-minor, matching the global flat row index order).
    // The first barrier here also guarantees every lane finished reading the
    // input tile before anyone overwrites it as the output stage.
    scan[t] = cnt;
    __syncthreads();
    for (int off = 1; off < PIX_PER_BLOCK; off <<= 1) {
        int v = scan[t];
        int add = (t >= off) ? scan[t - off] : 0;
        __syncthreads();
        scan[t] = v + add;
        __syncthreads();
    }
    int pv = scan[t] - cnt;                      // valid rows before this thread's rows
    const int cntTotal = scan[PIX_PER_BLOCK - 1];  // valid rows in this block

    // --- Compact decoded rows into LDS: valid packed first, zeros behind ----
    float2* stage2 = reinterpret_cast<float2*>(tile);
#pragma unroll
    for (int a = 0; a < AA; ++a) {
        const int lr = t * AA + a;                       // local row id (stable order)
        const int pos = flag[a] ? pv : (cntTotal + (lr - pv));
        float2* srow = stage2 + pos * 3;                 // 24 B per row
        if (flag[a]) {
            srow[0] = make_float2(conf[a], cx[a]);
            srow[1] = make_float2(cy[a], wv[a]);
            srow[2] = make_float2(hv[a], tv[a]);
        } else {
            float2 z = make_float2(0.0f, 0.0f);
            srow[0] = z; srow[1] = z; srow[2] = z;
        }
        pv += flag[a];
    }
    __syncthreads();

    // --- CDNA5 async LDS->global drain of the two contiguous spans ----------
    {
        const unsigned stageBase = (unsigned)(uintptr_t)(&tile[0]);
        const int vChunks = cntTotal * 3;                    // 8 B chunks, valid span
        const int iChunks = (ROWS_PER_BLOCK - cntTotal) * 3; // 8 B chunks, zero span

        // Valid span: rows [blockOff, blockOff+cntTotal) -> fully coalesced b64.
        const float* gv = out + (long long)blockOff * 6;     // uniform -> SGPR pair
        for (int k = t; k < vChunks; k += PIX_PER_BLOCK) {
            unsigned la   = stageBase + (unsigned)k * 8u;
            unsigned voff = (unsigned)k * 8u;
            asm volatile("global_store_async_from_lds_b64 %0, %1, %2"
                         :
                         : "v"(voff), "v"(la), "s"(gv)
                         : "memory");
        }

        // Zero span: rows [K + b*768 - blockOff, ...) (invalid rows before this
        // block = rows-before-block minus valid-before-block).
        const long long invBaseRow =
            (long long)K + ((long long)b * ROWS_PER_BLOCK - (long long)blockOff);
        const float* gi = out + invBaseRow * 6;              // uniform -> SGPR pair
        const unsigned stageInv = stageBase + (unsigned)vChunks * 8u;
        for (int k = t; k < iChunks; k += PIX_PER_BLOCK) {
            unsigned la   = stageInv + (unsigned)k * 8u;
            unsigned voff = (unsigned)k * 8u;
            asm volatile("global_store_async_from_lds_b64 %0, %1, %2"
                         :
                         : "v"(voff), "v"(la), "s"(gi)
                         : "memory");
        }
        asm volatile("s_wait_asynccnt 0x0" ::: "memory");
    }
}

// ---------------------------------------------------------------------------
extern "C" void kernel_launch(void* const* d_in, const int* in_sizes, int n_in,
                              void* d_out, int out_size, void* d_ws, size_t ws_size,
                              hipStream_t stream) {
    (void)in_sizes; (void)n_in; (void)out_size; (void)ws_size;

    const float* in  = (const float*)d_in[0];
    const float* thr = (const float*)d_in[1];
    float* out = (float*)d_out;

    int* counts  = (int*)d_ws;       // NB ints
    int* offsets = counts + NB;      // NB ints
    int* total   = offsets + NB;     // 1 int   (~51 KB of ws total)

    count_kernel <<<NB, PIX_PER_BLOCK, 0, stream>>>(in, thr, counts);
    scan_kernel  <<<1, 256, 0, stream>>>(counts, offsets, total);
    decode_kernel<<<NB, PIX_PER_BLOCK, 0, stream>>>(in, thr, offsets, total, out);
}